// QCNN_ZNOTY_29978871726315
// MI455X (gfx1250) — compile-verified
//
#include <hip/hip_runtime.h>
#include <math.h>
#include <stdint.h>

typedef __attribute__((ext_vector_type(2))) float v2f;
typedef __attribute__((ext_vector_type(8))) float v8f;

#define WAVES_PER_BLOCK 8
#define QBATCH 8192
// d_ws layout: 11 fixed 16x16 complex matrices, each 512 floats (256 re, 256 im)

struct cpx { float x, y; };
__device__ __forceinline__ cpx cmul(cpx a, cpx b){ return {a.x*b.x - a.y*b.y, a.x*b.y + a.y*b.x}; }
__device__ __forceinline__ cpx cadd(cpx a, cpx b){ return {a.x+b.x, a.y+b.y}; }

// ------------------------- setup kernel: build fixed 16x16 gate matrices -------------------------

__device__ void mk_rz(float t, cpx u[2][2]) {
  float s, c; __sincosf(0.5f*t, &s, &c);
  u[0][0]={c,-s}; u[0][1]={0,0}; u[1][0]={0,0}; u[1][1]={c,s};
}
__device__ void mk_ry(float t, cpx u[2][2]) {
  float s, c; __sincosf(0.5f*t, &s, &c);
  u[0][0]={c,0}; u[0][1]={-s,0}; u[1][0]={s,0}; u[1][1]={c,0};
}
__device__ void mk_u3(const float* p3, cpx u[2][2]) {
  float st, ct; __sincosf(0.5f*p3[0], &st, &ct);
  float sp, cp; __sincosf(p3[1], &sp, &cp);
  float sl, cl; __sincosf(p3[2], &sl, &cl);
  cpx ep={cp,sp}, el={cl,sl};
  u[0][0]={ct,0.f};
  u[0][1]={-el.x*st, -el.y*st};
  u[1][0]={ ep.x*st,  ep.y*st};
  cpx epl = cmul(ep, el);
  u[1][1]={epl.x*ct, epl.y*ct};
}
__device__ void mm4(cpx o[4][4], const cpx a[4][4], const cpx b[4][4]) {
  for (int i=0;i<4;++i) for (int j=0;j<4;++j) {
    cpx acc={0,0};
    for (int k=0;k<4;++k) acc = cadd(acc, cmul(a[i][k], b[k][j]));
    o[i][j]=acc;
  }
}
__device__ void apply4(cpx U[4][4], const cpx G[4][4]) {
  cpx T[4][4]; mm4(T,G,U);
  for (int i=0;i<4;++i) for (int j=0;j<4;++j) U[i][j]=T[i][j];
}
// 4x4 basis index = 2*b_lo + b_hi (lo = smaller wire number = more significant)
__device__ void g_hi(cpx G[4][4], const cpx u[2][2]) {
  for (int i=0;i<4;++i) for (int j=0;j<4;++j)
    G[i][j] = ((i>>1)==(j>>1)) ? u[i&1][j&1] : cpx{0,0};
}
__device__ void g_lo(cpx G[4][4], const cpx u[2][2]) {
  for (int i=0;i<4;++i) for (int j=0;j<4;++j)
    G[i][j] = ((i&1)==(j&1)) ? u[i>>1][j>>1] : cpx{0,0};
}
__device__ void g_cnot_hl(cpx G[4][4]) { // control hi (LSB), target lo (MSB)
  for (int i=0;i<4;++i) for (int j=0;j<4;++j)
    G[i][j] = (i == (((((j>>1)&1)^(j&1))<<1) | (j&1))) ? cpx{1,0} : cpx{0,0};
}
__device__ void g_cnot_lh(cpx G[4][4]) { // control lo (MSB), target hi (LSB)
  for (int i=0;i<4;++i) for (int j=0;j<4;++j)
    G[i][j] = (i == ((j&2) | ((j&1)^((j>>1)&1)))) ? cpx{1,0} : cpx{0,0};
}
// conv_block as a 4x4: rz(hi,a); CNOT(h,l); rz(lo,b); ry(hi,ya); CNOT(l,h); ry(hi,yb); CNOT(h,l); rz(lo,c)
__device__ void conv4(cpx U[4][4], float rza, float rzb, float rzc, float rya, float ryb) {
  for (int i=0;i<4;++i) for (int j=0;j<4;++j) U[i][j] = (i==j)?cpx{1,0}:cpx{0,0};
  cpx u[2][2], G[4][4];
  mk_rz(rza,u); g_hi(G,u); apply4(U,G);
  g_cnot_hl(G);            apply4(U,G);
  mk_rz(rzb,u); g_lo(G,u); apply4(U,G);
  mk_ry(rya,u); g_hi(G,u); apply4(U,G);
  g_cnot_lh(G);            apply4(U,G);
  mk_ry(ryb,u); g_hi(G,u); apply4(U,G);
  g_cnot_hl(G);            apply4(U,G);
  mk_rz(rzc,u); g_lo(G,u); apply4(U,G);
}
__device__ void store16(float* ws, int idx, const cpx M[16][16], int transpose) {
  float* re = ws + idx*512; float* im = re + 256;
  for (int r=0;r<16;++r) for (int k=0;k<16;++k) {
    cpx v = transpose ? M[k][r] : M[r][k];
    re[r*16+k]=v.x; im[r*16+k]=v.y;
  }
}

__global__ void qcnn_setup(const float* __restrict__ rz, const float* __restrict__ ry,
                           const float* __restrict__ u3p, float* __restrict__ ws) {
  if (threadIdx.x != 0 || blockIdx.x != 0) return;
  cpx M[16][16], A4[4][4], B4[4][4];
  // idx0: conv layer1 row = block(1,0) [bits3,2] x block(3,2) [bits1,0]
  conv4(A4, rz[0],rz[1],rz[2], ry[0],ry[1]);
  conv4(B4, rz[3],rz[4],rz[5], ry[2],ry[3]);
  for (int r=0;r<16;++r) for (int k=0;k<16;++k)
    M[r][k] = cmul(A4[r>>2][k>>2], B4[r&3][k&3]);
  store16(ws,0,M,0);
  // idx1: conv layer1 col = block(5,4) x block(7,6)  (stored transposed)
  conv4(A4, rz[6],rz[7],rz[8],  ry[4],ry[5]);
  conv4(B4, rz[9],rz[10],rz[11],ry[6],ry[7]);
  for (int r=0;r<16;++r) for (int k=0;k<16;++k)
    M[r][k] = cmul(A4[r>>2][k>>2], B4[r&3][k&3]);
  store16(ws,1,M,1);
  // idx2: u3_0 on wire1 (bit2), u3_1 on wire3 (bit0)
  cpx U0[2][2], U1[2][2];
  mk_u3(u3p+0,U0); mk_u3(u3p+3,U1);
  for (int r=0;r<16;++r) for (int k=0;k<16;++k)
    M[r][k] = (((r^k)&0xA)==0) ? cmul(U0[(r>>2)&1][(k>>2)&1], U1[r&1][k&1]) : cpx{0,0};
  store16(ws,2,M,0);
  // idx3: u3_2 on wire5 (bit2), u3_3 on wire7 (bit0)  (transposed)
  mk_u3(u3p+6,U0); mk_u3(u3p+9,U1);
  for (int r=0;r<16;++r) for (int k=0;k<16;++k)
    M[r][k] = (((r^k)&0xA)==0) ? cmul(U0[(r>>2)&1][(k>>2)&1], U1[r&1][k&1]) : cpx{0,0};
  store16(ws,3,M,1);
  // idx4: block(3,1): wires 1(bit2),3(bit0); 4x4 index = 2*b1 + b3
  conv4(A4, rz[12],rz[13],rz[14], ry[8],ry[9]);
  for (int r=0;r<16;++r) for (int k=0;k<16;++k)
    M[r][k] = (((r^k)&0xA)==0) ? A4[(((r>>2)&1)<<1)|(r&1)][(((k>>2)&1)<<1)|(k&1)] : cpx{0,0};
  store16(ws,4,M,0);
  // idx5: RZ(rz18 on wire5) * block(7,5): wires 5(bit2),7(bit0)  (transposed)
  conv4(A4, rz[15],rz[16],rz[17], ry[10],ry[11]);
  cpx rz5[2][2]; mk_rz(rz[18], rz5);
  for (int c=0;c<16;++c) for (int k=0;k<16;++k) {
    cpx v = (((c^k)&0xA)==0) ? A4[(((c>>2)&1)<<1)|(c&1)][(((k>>2)&1)<<1)|(k&1)] : cpx{0,0};
    M[c][k] = cmul(rz5[(c>>2)&1][(c>>2)&1], v);
  }
  store16(ws,5,M,1);
  // idx6: diagonal RZ(rz19) on wire3 (row bit0)
  cpx rz3[2][2]; mk_rz(rz[19], rz3);
  for (int r=0;r<16;++r) for (int k=0;k<16;++k)
    M[r][k] = (r==k) ? rz3[r&1][r&1] : cpx{0,0};
  store16(ws,6,M,0);
  // idx7 / idx8: RY(ry12)/RY(ry13) on wire5 (col bit2)  (transposed)
  cpx ryu[2][2];
  mk_ry(ry[12], ryu);
  for (int c=0;c<16;++c) for (int k=0;k<16;++k)
    M[c][k] = (((c^k)&0xB)==0) ? ryu[(c>>2)&1][(k>>2)&1] : cpx{0,0};
  store16(ws,7,M,1);
  mk_ry(ry[13], ryu);
  for (int c=0;c<16;++c) for (int k=0;k<16;++k)
    M[c][k] = (((c^k)&0xB)==0) ? ryu[(c>>2)&1][(k>>2)&1] : cpx{0,0};
  store16(ws,8,M,1);
  // idx9: (u3_4 * RZ(rz20)) on wire3 (row bit0)
  cpx U4[2][2]; mk_u3(u3p+12, U4);
  cpx rzc2[2][2]; mk_rz(rz[20], rzc2);
  cpx w22[2][2];
  for (int i=0;i<2;++i) for (int j=0;j<2;++j) w22[i][j] = cmul(U4[i][j], rzc2[j][j]);
  for (int r=0;r<16;++r) for (int k=0;k<16;++k)
    M[r][k] = (((r^k)&0xE)==0) ? w22[r&1][k&1] : cpx{0,0};
  store16(ws,9,M,0);
  // idx10: u3_5 on wire7 (col bit0)  (transposed)
  cpx U5[2][2]; mk_u3(u3p+15, U5);
  for (int c=0;c<16;++c) for (int k=0;k<16;++k)
    M[c][k] = (((c^k)&0xE)==0) ? U5[c&1][k&1] : cpx{0,0};
  store16(ws,10,M,1);
}

// ------------------------- main kernel: one wave32 per batch element -------------------------
// State P(r,c): r = wires 0-3 (wire0 = MSB), c = wires 4-7. Kept in LDS:
//   S[r*16+c] = Re, S[256 + r*16+c] = Im.
// WMMA f32 16x16x4 operand layouts (wave32):
//   A (16x4):  element (m=lane%16, k) ; VGPR pair per k-chunk, k = 4j + (reg&1) + 2*(lane>=16)
//   B (4x16):  element (k, n=lane%16) ; same k striping
//   C/D:       element (m = v + 8*(lane>=16), n=lane%16)

__device__ __forceinline__ v8f wmma4(float a0, float a1, float b0, float b1, v8f c) {
  v2f A; A.x=a0; A.y=a1;
  v2f B; B.x=b0; B.y=b1;
  return __builtin_amdgcn_wmma_f32_16x16x4_f32(false, A, false, B, (short)0, c, false, false);
}

// complex GEMM: D = A x B, 16 WMMAs across 4 independent accumulation chains
__device__ __forceinline__ void cgemm(const float* aR, const float* aI,
                                      const float* bR, const float* bI,
                                      v8f& Dr, v8f& Di) {
  v8f crr = {}, cii = {}, cri = {}, cir = {};
#pragma unroll
  for (int j = 0; j < 4; ++j) {
    crr = wmma4(aR[2*j], aR[2*j+1], bR[2*j], bR[2*j+1], crr);
    cii = wmma4(aI[2*j], aI[2*j+1], bI[2*j], bI[2*j+1], cii);
    cri = wmma4(aR[2*j], aR[2*j+1], bI[2*j], bI[2*j+1], cri);
    cir = wmma4(aI[2*j], aI[2*j+1], bR[2*j], bR[2*j+1], cir);
  }
  Dr = crr - cii;
  Di = cri + cir;
}

__device__ __forceinline__ void load_B_state(const float* S, int n, int hi, float* bR, float* bI) {
#pragma unroll
  for (int t=0;t<8;++t){ int k = 4*(t>>1)+(t&1)+2*hi; bR[t]=S[k*16+n]; bI[t]=S[256+k*16+n]; }
}
__device__ __forceinline__ void load_A_state(const float* S, int n, int hi, float* aR, float* aI) {
#pragma unroll
  for (int t=0;t<8;++t){ int k = 4*(t>>1)+(t&1)+2*hi; aR[t]=S[n*16+k]; aI[t]=S[256+n*16+k]; }
}
__device__ __forceinline__ void load_A_fixed(const float* g, int n, int hi, float* aR, float* aI) {
#pragma unroll
  for (int t=0;t<8;++t){ int k = 4*(t>>1)+(t&1)+2*hi; aR[t]=g[n*16+k]; aI[t]=g[256+n*16+k]; }
}
__device__ __forceinline__ void load_B_fixed(const float* g, int n, int hi, float* bR, float* bI) {
#pragma unroll
  for (int t=0;t<8;++t){ int k = 4*(t>>1)+(t&1)+2*hi; bR[t]=g[k*16+n]; bI[t]=g[256+k*16+n]; }
}
__device__ __forceinline__ void store_D(float* S, int n, int hi, v8f Dr, v8f Di) {
#pragma unroll
  for (int v=0; v<8; ++v){ int m=v+8*hi; S[m*16+n]=Dr[v]; S[256+m*16+n]=Di[v]; }
}

__device__ __forceinline__ void pass_left(float* S, int n, int hi, const float* aR, const float* aI) {
  float bR[8], bI[8]; load_B_state(S,n,hi,bR,bI);
  v8f Dr, Di; cgemm(aR,aI,bR,bI,Dr,Di);
  store_D(S,n,hi,Dr,Di);
}
__device__ __forceinline__ void pass_left_fixed(float* S, int n, int hi, const float* g) {
  float aR[8],aI[8]; load_A_fixed(g,n,hi,aR,aI);
  pass_left(S,n,hi,aR,aI);
}
__device__ __forceinline__ void pass_right(float* S, int n, int hi, const float* bR, const float* bI) {
  float aR[8],aI[8]; load_A_state(S,n,hi,aR,aI);
  v8f Dr,Di; cgemm(aR,aI,bR,bI,Dr,Di);
  store_D(S,n,hi,Dr,Di);
}
__device__ __forceinline__ void pass_right_fixed(float* S, int n, int hi, const float* g) {
  float bR[8],bI[8]; load_B_fixed(g,n,hi,bR,bI);
  pass_right(S,n,hi,bR,bI);
}

// KIND 0: fused per-cycle CNOT(3,4) -> CNOT(4,5)C(5,6)C(6,7) -> CNOT(7,0)
// KIND 1: CNOT(5,3)    KIND 2: CNOT(3,5)
template<int KIND>
__device__ __forceinline__ void perm_pass(float* S, int n, int hi) {
  float tr[8], ti[8];
#pragma unroll
  for (int v=0; v<8; ++v) {
    int m = v + 8*hi;
    int sr, sc;
    if (KIND==0)      { sr = m ^ ((n&1)<<3); sc = (n ^ (n>>1)) ^ ((m&1)<<3); }
    else if (KIND==1) { sr = m ^ ((n>>2)&1); sc = n; }
    else              { sr = m;              sc = n ^ ((m&1)<<2); }
    tr[v] = S[sr*16+sc]; ti[v] = S[256+sr*16+sc];
  }
#pragma unroll
  for (int v=0; v<8; ++v) { int m=v+8*hi; S[m*16+n]=tr[v]; S[256+m*16+n]=ti[v]; }
}

// per-wire factor of (RZ(phi)*RY(theta))^{x4}: element D4(b,k)
__device__ __forceinline__ void d4elem(int bb, int k, float c, float s, float ph, float& re, float& im) {
  float mag = 1.f;
#pragma unroll
  for (int w = 0; w < 4; ++w) {
    int bw = (bb>>w)&1, kw = (k>>w)&1;
    mag *= (bw==kw) ? c : (bw ? s : -s);
  }
  float t = ph * (float)(__popc(bb) - 2);
  float sn, cn; __sincosf(t, &sn, &cn);
  re = mag*cn; im = mag*sn;
}

// gfx1250 async global->LDS copy of one 16B chunk (tracked by ASYNCcnt)
__device__ __forceinline__ void async_copy_b128(uint32_t lds_off, const void* gptr) {
  asm volatile("global_load_async_to_lds_b128 %0, %1, off"
               :: "v"(lds_off), "v"((unsigned long long)(uintptr_t)gptr)
               : "memory");
}

__global__ void __launch_bounds__(32*WAVES_PER_BLOCK)
qcnn_main(const float* __restrict__ theta, const float* __restrict__ phi,
          const float* __restrict__ ws,
          const float* __restrict__ w1, const float* __restrict__ b1,
          const float* __restrict__ w2, const float* __restrict__ b2,
          float* __restrict__ out) {
  __shared__ float sh[WAVES_PER_BLOCK * 512];
  __shared__ float mats[11 * 512];           // 22528 B fixed gate-matrix table
  const int lane = threadIdx.x & 31;
  const int wv   = threadIdx.x >> 5;
  const int bidx = blockIdx.x * WAVES_PER_BLOCK + wv;
  float* S = sh + wv * 512;
  const int n  = lane & 15;
  const int hi = lane >> 4;

  // stage the fixed gate-matrix table into LDS once per workgroup via async DMA
  {
    const uint32_t mbase = (uint32_t)(uintptr_t)&mats[0];
    const char* gsrc = (const char*)ws;
#pragma unroll
    for (int i = 0; i < 6; ++i) {
      int idx = i * 256 + (int)threadIdx.x;          // 1408 x 16B chunks
      if (idx < 1408)
        async_copy_b128(mbase + (uint32_t)idx * 16u, gsrc + (size_t)idx * 16);
    }
    asm volatile("s_wait_asynccnt 0" ::: "memory");
  }

  // init |0...0>: P(0,0)=1
#pragma unroll
  for (int v = 0; v < 8; ++v) {
    int m = v + 8*hi;
    S[m*16+n]       = (m==0 && n==0) ? 1.0f : 0.0f;
    S[256 + m*16+n] = 0.0f;
  }

  // batched per-wave matrices from theta/phi
  const float th = theta[bidx], ph = phi[bidx];
  float sth, cth; __sincosf(0.5f*th, &sth, &cth);
  float mrR[8], mrI[8], mcR[8], mcI[8];
#pragma unroll
  for (int t = 0; t < 8; ++t) {
    int k = 4*(t>>1) + (t&1) + 2*hi;
    int bmu = n ^ (n>>1);                       // row-CNOT-chain permutation mu(m), m = lane%16
    d4elem(bmu, k, cth, sth, ph, mrR[t], mrI[t]);  // A-layout: M_row = Crow * D4
    d4elem(n,   k, cth, sth, ph, mcR[t], mcI[t]);  // B-layout: (D4_col)^T
  }

  __syncthreads();   // table visible to all waves

  // 4 cycles: RY^8, RZ^8, CNOT ring
  for (int cyc = 0; cyc < 4; ++cyc) {
    pass_left (S,n,hi, mrR, mrI);   // rows: Crow * (RZ*RY)^{x4}
    pass_right(S,n,hi, mcR, mcI);   // cols: (RZ*RY)^{x4}
    perm_pass<0>(S,n,hi);           // CNOT(3,4), col CNOTs, CNOT(7,0)
  }
  // conv layer 1
  pass_left_fixed (S,n,hi, mats + 0*512);
  pass_right_fixed(S,n,hi, mats + 1*512);
  // u3 on wires 1,3,5,7
  pass_left_fixed (S,n,hi, mats + 2*512);
  pass_right_fixed(S,n,hi, mats + 3*512);
  // conv layer 2: block(3,1), [rz5a * block(7,5)], then block(5,3) decomposition
  pass_left_fixed (S,n,hi, mats + 4*512);
  pass_right_fixed(S,n,hi, mats + 5*512);
  perm_pass<1>(S,n,hi);                       // CNOT(5,3)
  pass_left_fixed (S,n,hi, mats + 6*512);     // RZ(rzb) on wire3
  pass_right_fixed(S,n,hi, mats + 7*512);     // RY(rya) on wire5
  perm_pass<2>(S,n,hi);                       // CNOT(3,5)
  pass_right_fixed(S,n,hi, mats + 8*512);     // RY(ryb) on wire5
  perm_pass<1>(S,n,hi);                       // CNOT(5,3)
  pass_left_fixed (S,n,hi, mats + 9*512);     // u3_4 * RZ(rzc) on wire3

  // final col pass (u3_5 on wire7), keep result in registers for readout
  float bR[8], bI[8], aR[8], aI[8];
  load_B_fixed(mats + 10*512, n, hi, bR, bI);
  load_A_state(S, n, hi, aR, aI);
  v8f Dr, Di; cgemm(aR, aI, bR, bI, Dr, Di);

  // expval_z(wire3) -> sign by row bit0 (= v&1); expval_z(wire7) -> sign by col bit0 (= lane&1)
  float acc0 = 0.f, acc1 = 0.f;
#pragma unroll
  for (int v = 0; v < 8; ++v) {
    float p = Dr[v]*Dr[v] + Di[v]*Di[v];
    acc0 += (v & 1) ? -p : p;
    acc1 += p;
  }
  acc1 = (lane & 1) ? -acc1 : acc1;
#pragma unroll
  for (int off = 16; off > 0; off >>= 1) {
    acc0 += __shfl_xor(acc0, off, 32);
    acc1 += __shfl_xor(acc1, off, 32);
  }
  if (lane == 0) {
    float x0 = acc0, x1 = acc1;
    float y = b2[0];
    for (int j = 0; j < 10; ++j) {
      float h = tanhf(w1[2*j]*x0 + w1[2*j+1]*x1 + b1[j]);
      y += w2[j]*h;
    }
    out[bidx] = 1.0f / (1.0f + __expf(-y));
  }
}

// ------------------------- launch -------------------------

extern "C" void kernel_launch(void* const* d_in, const int* in_sizes, int n_in,
                              void* d_out, int out_size, void* d_ws, size_t ws_size,
                              hipStream_t stream) {
  const float* theta = (const float*)d_in[0];
  const float* phi   = (const float*)d_in[1];
  const float* rz    = (const float*)d_in[2];
  const float* ry    = (const float*)d_in[3];
  const float* u3p   = (const float*)d_in[4];
  const float* w1    = (const float*)d_in[5];
  const float* b1    = (const float*)d_in[6];
  const float* w2    = (const float*)d_in[7];
  const float* b2    = (const float*)d_in[8];
  float* ws = (float*)d_ws;

  qcnn_setup<<<1, 32, 0, stream>>>(rz, ry, u3p, ws);
  qcnn_main<<<QBATCH / WAVES_PER_BLOCK, 32 * WAVES_PER_BLOCK, 0, stream>>>(
      theta, phi, ws, w1, b1, w2, b2, (float*)d_out);
}